// QuadraticProjectionLoss_39513699123327
// MI455X (gfx1250) — compile-verified
//
#include <hip/hip_runtime.h>
#include <math.h>

typedef float v2f __attribute__((ext_vector_type(2)));
typedef float v8f __attribute__((ext_vector_type(8)));

#define VV 8
#define JJ 32
#define SCALE_KPS 0.1f
#define THRESH 100.0f
#define POW100_09 63.095734448019324f   // 100^0.9

__device__ __forceinline__ float pow01_fast(float x) {
    // x^0.1 for x > 100 (safe domain for native log2/exp2)
#if __has_builtin(__builtin_amdgcn_logf) && __has_builtin(__builtin_amdgcn_exp2f)
    return __builtin_amdgcn_exp2f(0.1f * __builtin_amdgcn_logf(x));
#else
    return __powf(x, 0.1f);
#endif
}

// One block (128 threads = 4 waves) per batch b.
// Wave w handles the view pair (2w, 2w+1): lanes 0-15 -> view 2w, lanes 16-31 -> view 2w+1.
// WMMA A (16x4): rows 0-2 = M(view 2w), rows 8-10 = M(view 2w+1), rest zero.
// WMMA B (4x16): column n = [kx ky kz 1]^T of joint t*16+n.
// D: every lane reads (x,y,z) of joint t*16+(lane&15) of its own view in D[0..2].
__global__ __launch_bounds__(128) void proj_loss_kernel(
    const float* __restrict__ Kmat,
    const float* __restrict__ cam,
    const float* __restrict__ kps,
    const float* __restrict__ ikps,
    float* __restrict__ partial)
{
    const int b    = blockIdx.x;
    const int lane = threadIdx.x & 31;
    const int wp   = threadIdx.x >> 5;       // view-pair id 0..3
    const int n    = lane & 15;
    const int h    = lane >> 4;              // 0: cols {0,1}; 1: cols {2,3}
    const int v    = 2 * wp + h;             // this lane's view for the loss phase

    // ---- A operand (branchless): lane supplies {A[r][2h], A[r][2h+1]}, r = n ----
    // rows 0-2 from view 2wp+0, rows 8-10 from view 2wp+1, others zero.
    const bool valid = (n < 3) || (n >= 8 && n < 11);
    const int  mrow  = valid ? (n & 7)  : 0;   // clamped: all loads stay in-bounds
    const int  mview = valid ? (n >> 3) : 0;
    float ax, ay;
    {
        const size_t mb = (size_t)b * VV + 2 * wp + mview;
        const float* Kp = Kmat + mb * 9 + (size_t)mrow * 3;  // K row
        const float* Cp = cam  + mb * 12 + 2 * h;            // cam 3x4, cols 2h,2h+1
        const float k0 = Kp[0], k1 = Kp[1], k2 = Kp[2];
        const float axv = k0 * Cp[0] + k1 * Cp[4] + k2 * Cp[8];
        const float ayv = k0 * Cp[1] + k1 * Cp[5] + k2 * Cp[9];
        ax = valid ? axv : 0.0f;
        ay = valid ? ayv : 0.0f;
    }
    const v2f A = {ax, ay};

    // ---- initial keypoints: this lane owns joints n and n+16 of view v ----
    const size_t base_v = (size_t)b * VV + v;
    const float2* ikp2 = reinterpret_cast<const float2*>(ikps);
    const float2 ik0 = ikp2[base_v * JJ + n];
    const float2 ik1 = ikp2[base_v * JJ + n + 16];
    float s_init = ik0.x*ik0.x + ik0.y*ik0.y + ik1.x*ik1.x + ik1.y*ik1.y;

    float s_proj = 0.0f, s_loss = 0.0f;

#pragma unroll
    for (int t = 0; t < 2; ++t) {
        // B operand (branchless): lane n -> {kx,ky}, lane 16+n -> {kz,1}
        const float* kw = kps + ((size_t)b * JJ + t * 16 + n) * 3;
        const float b0  = kw[2 * h];           // kx (lo) / kz (hi)
        const float kw1 = kw[1];               // always in-bounds, all lanes
        const v2f Bv = {b0, h ? 1.0f : kw1};

        const v8f Cv = {0.f, 0.f, 0.f, 0.f, 0.f, 0.f, 0.f, 0.f};
        const v8f D = __builtin_amdgcn_wmma_f32_16x16x4_f32(
            false, A, false, Bv, (short)0, Cv, false, false);

        // lanes 0-15 read rows 0-2 (view 2wp), lanes 16-31 rows 8-10 (view 2wp+1)
        const float invz = __builtin_amdgcn_rcpf(D[2]);
        const float px = D[0] * invz;
        const float py = D[1] * invz;
        const float ixx = t ? ik1.x : ik0.x;
        const float iyy = t ? ik1.y : ik0.y;
        s_proj += px * px + py * py;

        float dx = (px - ixx) * SCALE_KPS; dx = dx * dx;
        float dy = (py - iyy) * SCALE_KPS; dy = dy * dy;
        const float dxs = pow01_fast(dx) * POW100_09;
        const float dys = pow01_fast(dy) * POW100_09;
        dx = (dx > THRESH) ? dxs : dx;
        dy = (dy > THRESH) ? dys : dy;
        s_loss += dx + dy;
    }

    // ---- per-view butterfly reduction within each 16-lane half ----
#pragma unroll
    for (int off = 8; off >= 1; off >>= 1) {
        s_proj += __shfl_xor(s_proj, off, 32);
        s_init += __shfl_xor(s_init, off, 32);
        s_loss += __shfl_xor(s_loss, off, 32);
    }

    __shared__ float sm[VV];
    if (n == 0) {  // lanes 0 and 16 of each wave: one view each
        const float penal = fabsf(sqrtf(s_proj) * __builtin_amdgcn_rcpf(sqrtf(s_init)) - 1.0f);
        sm[v] = (s_loss * 0.5f) * penal;
    }
    __syncthreads();
    if (threadIdx.x == 0) {
        float s = 0.0f;
#pragma unroll
        for (int i = 0; i < VV; ++i) s += sm[i];
        partial[b] = s;
    }
}

__global__ __launch_bounds__(256) void reduce_kernel(
    const float* __restrict__ p, float* __restrict__ out, int nblk, float scale)
{
    __shared__ float sm[256];
    float s = 0.0f;
    for (int i = threadIdx.x; i < nblk; i += 256) s += p[i];
    sm[threadIdx.x] = s;
    __syncthreads();
    for (int w = 128; w > 0; w >>= 1) {
        if ((int)threadIdx.x < w) sm[threadIdx.x] += sm[threadIdx.x + w];
        __syncthreads();
    }
    if (threadIdx.x == 0) out[0] = sm[0] * scale;
}

extern "C" void kernel_launch(void* const* d_in, const int* in_sizes, int n_in,
                              void* d_out, int out_size, void* d_ws, size_t ws_size,
                              hipStream_t stream) {
    const float* Kmat = (const float*)d_in[0];  // [B,V,3,3]
    const float* cam  = (const float*)d_in[1];  // [B,V,3,4]
    const float* kps  = (const float*)d_in[2];  // [B,J,3]
    const float* ikps = (const float*)d_in[3];  // [B,V,J,2]
    const int Bn = in_sizes[3] / (VV * JJ * 2);

    float* partial = (float*)d_ws;              // Bn floats of scratch

    proj_loss_kernel<<<Bn, 128, 0, stream>>>(Kmat, cam, kps, ikps, partial);
    reduce_kernel<<<1, 256, 0, stream>>>(partial, (float*)d_out, Bn,
                                         1.0f / (float)(Bn * VV));
}